// GCN_51264729645358
// MI455X (gfx1250) — compile-verified
//
#include <hip/hip_runtime.h>
#include <hip/hip_bf16.h>

// ---------------------------------------------------------------------------
// GCN on dynamic thresholded-cosine graph, M=8192 nodes, C=64.
// Adjacency kept as an 8 MB bitmask (L2-resident). All GEMM-like work uses
// V_WMMA_F32_16X16X4_F32. Tile staging uses GLOBAL_LOAD_ASYNC_TO_LDS_B128
// (ASYNCcnt) with LDS double-buffering so the 4 KB j-tile fetch overlaps the
// 16 WMMAs of the previous tile.
// ---------------------------------------------------------------------------

typedef float v2f __attribute__((ext_vector_type(2)));
typedef float v8f __attribute__((ext_vector_type(8)));

#define M_NODES   8192
#define C_DIM     64
#define TILE      16
#define NTILES    (M_NODES / TILE)   // 512
#define THR       0.85f
#define LAMBDA_   0.5f
#define EPS_      1e-12f

// workspace layout (bytes)
#define OFF_XN    0u                                  // 8192*64*4  = 2 MB
#define OFF_DINV  (OFF_XN   + M_NODES * C_DIM * 4u)   // 8192*4     = 32 KB
#define OFF_MASK  (OFF_DINV + M_NODES * 4u)           // 512*512*8*4= 8 MB
#define OFF_YD    (OFF_MASK + NTILES * NTILES * 8u * 4u)
#define OFF_H     (OFF_YD   + M_NODES * C_DIM * 4u)

static __device__ __forceinline__ v8f wmma_f32(v2f a, v2f b, v8f c) {
  return __builtin_amdgcn_wmma_f32_16x16x4_f32(false, a, false, b, (short)0, c,
                                               false, false);
}

// async copy 16 B/lane: global -> LDS (tracked by ASYNCcnt)
static __device__ __forceinline__ void async_b128(unsigned lds_off,
                                                  const float* gaddr) {
  asm volatile("global_load_async_to_lds_b128 %0, %1, off"
               :: "v"(lds_off), "v"(gaddr)
               : "memory");
}
static __device__ __forceinline__ void wait_async0() {
  asm volatile("s_wait_asynccnt 0x0" ::: "memory");
}

// --- kernel 1: row L2-normalize ------------------------------------------
__global__ void k_normalize(const float* __restrict__ x, float* __restrict__ xn) {
  int row  = blockIdx.x * (blockDim.x >> 5) + (threadIdx.x >> 5);
  int lane = threadIdx.x & 31;
  float2 v = *(const float2*)(x + row * C_DIM + lane * 2);
  float s = v.x * v.x + v.y * v.y;
  #pragma unroll
  for (int off = 16; off > 0; off >>= 1) s += __shfl_xor(s, off, 32);
  float inv = 1.0f / fmaxf(sqrtf(s), EPS_);
  float2 o; o.x = v.x * inv; o.y = v.y * inv;
  *(float2*)(xn + row * C_DIM + lane * 2) = o;
}

// --- kernel 2: sim tiles -> mask bits + degrees -> dinv -------------------
__global__ void k_simmask(const float* __restrict__ xn,
                          unsigned* __restrict__ maskp,
                          float* __restrict__ dinv) {
  __shared__ float smem[2][TILE * C_DIM];   // 2 x 4 KB j-tiles of xn
  const int lane = threadIdx.x & 31;
  const int wid  = threadIdx.x >> 5;
  const int hi   = lane >> 4;
  const int lo   = lane & 15;
  const int ti   = blockIdx.x * 8 + wid;
  const int rowA = ti * TILE + lo;

  const unsigned lds0 = (unsigned)(uintptr_t)(&smem[0][0]) + threadIdx.x * 16u;
  const unsigned lds1 = lds0 + TILE * C_DIM * 4u;

  // prologue: kick off tile 0 into buffer 0, then load A regs under it
  async_b128(lds0, xn + (size_t)threadIdx.x * 4);

  v2f a[16];
  #pragma unroll
  for (int kc = 0; kc < 16; ++kc) {
    float2 p = *(const float2*)(xn + rowA * C_DIM + kc * 4 + hi * 2);
    a[kc].x = p.x; a[kc].y = p.y;
  }

  unsigned deg = 0;  // degree of row (ti*16 + lo)

  for (int tj = 0; tj < NTILES; ++tj) {
    const int cur = tj & 1;
    wait_async0();          // my slice of tile tj landed in LDS
    __syncthreads();        // everyone's slice landed; buf cur^1 free to reuse
    if (tj + 1 < NTILES)    // overlap next fetch with this tile's WMMAs
      async_b128(cur ? lds0 : lds1,
                 xn + (size_t)(tj + 1) * TILE * C_DIM + threadIdx.x * 4);
    const float* sb = &smem[cur][0];

    v8f c = {0.f, 0.f, 0.f, 0.f, 0.f, 0.f, 0.f, 0.f};
    #pragma unroll
    for (int kc = 0; kc < 16; ++kc) {
      float2 b2 = *(const float2*)(sb + lo * C_DIM + kc * 4 + hi * 2);
      v2f b; b.x = b2.x; b.y = b2.y;
      c = wmma_f32(a[kc], b, c);
    }

    unsigned mk[8];
    #pragma unroll
    for (int k = 0; k < 8; ++k)
      mk[k] = __builtin_amdgcn_ballot_w32(c[k] > THR);  // row k | row k+8

    unsigned mymask = 0;
    #pragma unroll
    for (int k = 0; k < 8; ++k) mymask = ((lo & 7) == k) ? mk[k] : mymask;
    deg += __popc((mymask >> ((lo >> 3) * 16)) & 0xFFFFu);

    if (lane == 0) {
      uint4* mt = (uint4*)(maskp + ((size_t)ti * NTILES + tj) * 8u);
      mt[0] = make_uint4(mk[0], mk[1], mk[2], mk[3]);
      mt[1] = make_uint4(mk[4], mk[5], mk[6], mk[7]);
    }
  }

  if (lane < 16)  // +1 for the explicit self-loop (mask diag already counted)
    dinv[ti * TILE + lo] = rsqrtf((float)(deg + 1u));
}

// --- kernel 3/5: Yd = dinv .* (Ain @ Wm) ---------------------------------
__global__ void k_gemm_scale(const float* __restrict__ Ain,
                             const float* __restrict__ Wm,
                             const float* __restrict__ dinv,
                             float* __restrict__ Yd) {
  const int lane = threadIdx.x & 31;
  const int wid  = threadIdx.x >> 5;
  const int hi   = lane >> 4;
  const int lo   = lane & 15;
  const int ti   = blockIdx.x * 8 + wid;
  const int rowA = ti * TILE + lo;

  v2f a[16];
  #pragma unroll
  for (int kc = 0; kc < 16; ++kc) {
    float2 p = *(const float2*)(Ain + rowA * C_DIM + kc * 4 + hi * 2);
    a[kc].x = p.x; a[kc].y = p.y;
  }

  float dv[8];
  #pragma unroll
  for (int v = 0; v < 8; ++v) dv[v] = dinv[ti * TILE + v + 8 * hi];

  #pragma unroll
  for (int nc = 0; nc < 4; ++nc) {
    v8f c = {0.f, 0.f, 0.f, 0.f, 0.f, 0.f, 0.f, 0.f};
    #pragma unroll
    for (int kc = 0; kc < 16; ++kc) {
      v2f b;
      b.x = Wm[(kc * 4 + hi * 2 + 0) * C_DIM + nc * 16 + lo];
      b.y = Wm[(kc * 4 + hi * 2 + 1) * C_DIM + nc * 16 + lo];
      c = wmma_f32(a[kc], b, c);
    }
    #pragma unroll
    for (int v = 0; v < 8; ++v) {
      int r = ti * TILE + v + 8 * hi;
      Yd[r * C_DIM + nc * 16 + lo] = c[v] * dv[v];
    }
  }
}

// --- kernel 4/6: Z = dinv .* (mask @ Yd + Yd) ; epilogue per mode ---------
// mode 1: H = relu(Z + b)            mode 2: out = resid + 0.5*(Z + b)
__global__ void k_aggregate(const unsigned* __restrict__ maskp,
                            const float* __restrict__ Yd,
                            const float* __restrict__ dinv,
                            const float* __restrict__ bias,
                            const float* __restrict__ resid,
                            float* __restrict__ outp, int mode) {
  __shared__ float smem[2][TILE * C_DIM];   // 2 x 4 KB j-tiles of Yd
  const int lane = threadIdx.x & 31;
  const int wid  = threadIdx.x >> 5;
  const int hi   = lane >> 4;
  const int lo   = lane & 15;
  const int ti   = blockIdx.x * 8 + wid;

  const unsigned lds0 = (unsigned)(uintptr_t)(&smem[0][0]) + threadIdx.x * 16u;
  const unsigned lds1 = lds0 + TILE * C_DIM * 4u;

  async_b128(lds0, Yd + (size_t)threadIdx.x * 4);  // tile 0 -> buf 0

  v8f c[4];
  #pragma unroll
  for (int nc = 0; nc < 4; ++nc)
    c[nc] = (v8f){0.f, 0.f, 0.f, 0.f, 0.f, 0.f, 0.f, 0.f};

  for (int tj = 0; tj < NTILES; ++tj) {
    const int cur = tj & 1;
    wait_async0();
    __syncthreads();
    if (tj + 1 < NTILES)
      async_b128(cur ? lds0 : lds1,
                 Yd + (size_t)(tj + 1) * TILE * C_DIM + threadIdx.x * 4);
    const float* sb = &smem[cur][0];

    // this lane's adjacency row bits for (row=lo, cols tj*16..tj*16+15)
    unsigned mdw = maskp[((size_t)ti * NTILES + tj) * 8u + (lo & 7)];
    unsigned rowbits = (mdw >> ((lo >> 3) * 16)) & 0xFFFFu;

    v2f am[4];
    #pragma unroll
    for (int kc = 0; kc < 4; ++kc) {
      am[kc].x = (float)((rowbits >> (kc * 4 + 2 * hi + 0)) & 1u);
      am[kc].y = (float)((rowbits >> (kc * 4 + 2 * hi + 1)) & 1u);
    }

    #pragma unroll
    for (int nc = 0; nc < 4; ++nc) {
      #pragma unroll
      for (int kc = 0; kc < 4; ++kc) {
        v2f b;
        b.x = sb[(kc * 4 + 2 * hi + 0) * C_DIM + nc * 16 + lo];
        b.y = sb[(kc * 4 + 2 * hi + 1) * C_DIM + nc * 16 + lo];
        c[nc] = wmma_f32(am[kc], b, c[nc]);
      }
    }
  }

  float dv[8];
  #pragma unroll
  for (int v = 0; v < 8; ++v) dv[v] = dinv[ti * TILE + v + 8 * hi];

  #pragma unroll
  for (int nc = 0; nc < 4; ++nc) {
    int col = nc * 16 + lo;
    float bc = bias[col];
    #pragma unroll
    for (int v = 0; v < 8; ++v) {
      int r = ti * TILE + v + 8 * hi;
      float z = dv[v] * (c[nc][v] + Yd[r * C_DIM + col]) + bc;
      if (mode == 1) {
        outp[r * C_DIM + col] = fmaxf(z, 0.0f);
      } else {
        outp[r * C_DIM + col] = resid[r * C_DIM + col] + LAMBDA_ * z;
      }
    }
  }
}

// ---------------------------------------------------------------------------
extern "C" void kernel_launch(void* const* d_in, const int* in_sizes, int n_in,
                              void* d_out, int out_size, void* d_ws, size_t ws_size,
                              hipStream_t stream) {
  const float* x  = (const float*)d_in[0];
  const float* W1 = (const float*)d_in[1];
  const float* b1 = (const float*)d_in[2];
  const float* W2 = (const float*)d_in[3];
  const float* b2 = (const float*)d_in[4];
  float* out = (float*)d_out;

  char* ws = (char*)d_ws;
  float*    xn   = (float*)(ws + OFF_XN);
  float*    dinv = (float*)(ws + OFF_DINV);
  unsigned* mask = (unsigned*)(ws + OFF_MASK);
  float*    yd   = (float*)(ws + OFF_YD);
  float*    hbuf = (float*)(ws + OFF_H);

  (void)in_sizes; (void)n_in; (void)out_size; (void)ws_size;

  k_normalize<<<M_NODES / 8, 256, 0, stream>>>(x, xn);
  k_simmask<<<NTILES / 8, 256, 0, stream>>>(xn, mask, dinv);
  k_gemm_scale<<<NTILES / 8, 256, 0, stream>>>(x, W1, dinv, yd);
  k_aggregate<<<NTILES / 8, 256, 0, stream>>>(mask, yd, dinv, b1, nullptr, hbuf, 1);
  k_gemm_scale<<<NTILES / 8, 256, 0, stream>>>(hbuf, W2, dinv, yd);
  k_aggregate<<<NTILES / 8, 256, 0, stream>>>(mask, yd, dinv, b2, x, out, 2);
}